// GaussianLayer_77120432767510
// MI455X (gfx1250) — compile-verified
//
#include <hip/hip_runtime.h>

typedef float v2f __attribute__((ext_vector_type(2)));
typedef float v8f __attribute__((ext_vector_type(8)));

#define B_DIM 512
#define F_DIM 784
#define R_DIM 512
#define C_DIM 16
#define D_DIM 49
#define KSTEPS 25          // K = 2*49 = 98, padded to 100 = 25 steps of 4
#define ROWSTRIDE 100      // floats per feature/coefficient row (k-extent)
#define BTILE 128          // b rows per workgroup = 8 waves * 16
#define HALF_LOG_2PI 0.9189385332046727f

__global__ __launch_bounds__(256)
void gaussian_wmma_kernel(const float* __restrict__ x,
                          const int*   __restrict__ mask,
                          const float* __restrict__ loc,
                          const float* __restrict__ scale,
                          float*       __restrict__ out)
{
    // LDS staging: A-features [x^2, x] interleaved along K, B-coefs [a2, a1].
    __shared__ float feat[BTILE * ROWSTRIDE];    // 51.2 KB
    __shared__ float coef[C_DIM * ROWSTRIDE];    //  6.4 KB
    __shared__ float a0tmp[C_DIM * D_DIM];       //  3.1 KB
    __shared__ float a0sum[C_DIM];
    __shared__ int   mrow[D_DIM];

    const int t    = threadIdx.x;    // 0..255
    const int wave = t >> 5;         // 0..7
    const int lane = t & 31;
    const int half = lane >> 4;      // which 16-lane half of the wave32
    const int l16  = lane & 15;

    const int r  = blockIdx.x >> 2;             // gridDim.x = R * (B/BTILE)
    const int b0 = (blockIdx.x & 3) * BTILE;

    // Phase 0: cache this r's mask row.
    if (t < D_DIM) mrow[t] = mask[r * D_DIM + t];
    __syncthreads();

    // Phase 1: coefficients a2 = -1/(2s^2), a1 = mu/s^2, per-entry a0.
    for (int e = t; e < C_DIM * D_DIM; e += 256) {
        const int c = e / D_DIM;
        const int d = e - c * D_DIM;
        const float lo  = loc  [(size_t)r * (C_DIM * D_DIM) + e];
        const float sc  = scale[(size_t)r * (C_DIM * D_DIM) + e];
        const float inv = 1.0f / (sc * sc);
        coef[c * ROWSTRIDE + 2 * d]     = -0.5f * inv;        // pairs with x^2
        coef[c * ROWSTRIDE + 2 * d + 1] = lo * inv;           // pairs with x
        a0tmp[e] = -0.5f * lo * lo * inv - __logf(sc) - HALF_LOG_2PI;
    }
    if (t < C_DIM) {                 // zero-pad K = 98,99
        coef[t * ROWSTRIDE + 98] = 0.0f;
        coef[t * ROWSTRIDE + 99] = 0.0f;
    }

    // Phase 2: gather x through the mask; build [x^2, x] features.
    {
        const int row = wave * 16 + l16;            // 0..127
        float* frow = &feat[row * ROWSTRIDE];
        const float* xrow = x + (size_t)(b0 + row) * F_DIM;
        for (int d = half; d < D_DIM; d += 2) {     // 2 lanes share a row
            const float xv = xrow[mrow[d]];
            frow[2 * d]     = xv * xv;
            frow[2 * d + 1] = xv;
        }
        if (half == 0) { frow[98] = 0.0f; frow[99] = 0.0f; }
    }
    __syncthreads();

    // Deterministic a0 reduction over d (no float atomics).
    if (t < C_DIM) {
        float s = 0.0f;
        for (int d = 0; d < D_DIM; ++d) s += a0tmp[t * D_DIM + d];
        a0sum[t] = s;
    }
    __syncthreads();

    // Phase 3: D = A(16x100) * B(100x16) + a0sum via V_WMMA_F32_16X16X4_F32.
    // f32 16x4 A layout: lanes 0-15 hold K=k0,k0+1; lanes 16-31 hold K=k0+2,k0+3.
    const int  arow  = wave * 16 + l16;
    const float* aPtr = &feat[arow * ROWSTRIDE + 2 * half];
    const float* bPtr = &coef[l16  * ROWSTRIDE + 2 * half];

    const float c0 = a0sum[l16];     // C/D layout: N = lane%16 in every VGPR
    v8f acc;
    #pragma unroll
    for (int i = 0; i < 8; ++i) acc[i] = c0;

    #pragma unroll
    for (int k = 0; k < KSTEPS; ++k) {
        v2f a = *(const v2f*)(aPtr + 4 * k);   // ds_load_b64, 8B aligned
        v2f b = *(const v2f*)(bPtr + 4 * k);
        acc = __builtin_amdgcn_wmma_f32_16x16x4_f32(
            /*neg_a=*/false, a, /*neg_b=*/false, b,
            /*c_mod=*/(short)0, acc, /*reuse_a=*/false, /*reuse_b=*/false);
    }

    // Phase 4: out[b][r][c]; VGPR v holds M = v + 8*half, N = l16.
    // For each v, lanes 0-15 write 64 contiguous bytes -> coalesced.
    float* obase = out + ((size_t)(b0 + wave * 16 + 8 * half) * R_DIM + r) * C_DIM + l16;
    #pragma unroll
    for (int v = 0; v < 8; ++v) {
        obase[(size_t)v * (R_DIM * C_DIM)] = acc[v];
    }
}

extern "C" void kernel_launch(void* const* d_in, const int* in_sizes, int n_in,
                              void* d_out, int out_size, void* d_ws, size_t ws_size,
                              hipStream_t stream) {
    (void)in_sizes; (void)n_in; (void)out_size; (void)d_ws; (void)ws_size;
    const float* x     = (const float*)d_in[0];
    const int*   mask  = (const int*)  d_in[1];
    const float* loc   = (const float*)d_in[2];
    const float* scale = (const float*)d_in[3];
    float*       out   = (float*)d_out;

    const int grid = R_DIM * (B_DIM / BTILE);   // 512 * 4 = 2048 workgroups
    gaussian_wmma_kernel<<<grid, 256, 0, stream>>>(x, mask, loc, scale, out);
}